// ReadUnit_13125420057184
// MI455X (gfx1250) — compile-verified
//
#include <hip/hip_runtime.h>
#include <hip/hip_bf16.h>

// dim=512, B=16, N=4096; alpha=1.5 => power = 1/(alpha-1) = 2 (exact square),
// d^(1-alpha) = 4096^-0.5 = 1/64.
#define DIM 512
#define NB  16
#define NN  4096

typedef float v2f __attribute__((ext_vector_type(2)));
typedef float v8f __attribute__((ext_vector_type(8)));

__device__ __forceinline__ float block_reduce_sum_256(float val, float* red) {
    const int t = threadIdx.x;
    __syncthreads();
    red[t] = val;
    __syncthreads();
    for (int s = 128; s > 0; s >>= 1) {
        if (t < s) red[t] += red[t + s];
        __syncthreads();
    }
    const float r = red[0];
    __syncthreads();
    return r;
}

// ---------------------------------------------------------------------------
// Kernel 1: the two small GEMMs (M = B = 16 rows) via V_WMMA_F32_16X16X4_F32.
//   tiles 0..31 : mm[b,d]  = sum_e m1[b,e] * Wm[d,e] + bm[d]      (N=512)
//   tiles 32..95: p[b,e]   = sum_d (c1[b,d]*Wa[d]) * Wc[d,e]      (N=1024)
// One wave (32 lanes) per 16x16 output tile, K stepped by 4 (128 WMMAs/tile).
// A layout (16x4 f32): lanes 0-15 -> K={0,1} in v[0],v[1]; lanes 16-31 -> K={2,3}.
// B layout mirrors A with N on lanes. C/D: VGPR i, lane L -> M=i+8*(L>>4), N=L&15.
// ---------------------------------------------------------------------------
__global__ __launch_bounds__(32) void gemm_small_wmma(
    const float* __restrict__ m1, const float* __restrict__ Wm,
    const float* __restrict__ bm, const float* __restrict__ c1,
    const float* __restrict__ Wa, const float* __restrict__ Wc,
    float* __restrict__ mm, float* __restrict__ p)
{
    const int tile = blockIdx.x;
    const int lane = threadIdx.x;
    const int mrow = lane & 15;
    const int half = lane >> 4;
    const int kb   = half * 2;
    v8f acc = {};

    if (tile < 32) {                       // mm = m1 @ Wm^T (+ bm)
        const int ncol = tile * 16 + (lane & 15);
        for (int kk = 0; kk < DIM; kk += 4) {
            const int k0 = kk + kb;
            v2f a, b;
            a.x = m1[mrow * DIM + k0];
            a.y = m1[mrow * DIM + k0 + 1];
            b.x = Wm[ncol * DIM + k0];         // B[k,n] = Wm[n,k]
            b.y = Wm[ncol * DIM + k0 + 1];
            acc = __builtin_amdgcn_wmma_f32_16x16x4_f32(
                false, a, false, b, (short)0, acc, false, false);
        }
        const float bias = bm[ncol];
        #pragma unroll
        for (int i = 0; i < 8; ++i)
            mm[(i + 8 * half) * DIM + ncol] = acc[i] + bias;
    } else {                               // p = (c1*Wa) @ Wc
        const int ncol = (tile - 32) * 16 + (lane & 15);
        for (int kk = 0; kk < DIM; kk += 4) {
            const int k0 = kk + kb;
            v2f a, b;
            a.x = c1[mrow * DIM + k0]     * Wa[k0];
            a.y = c1[mrow * DIM + k0 + 1] * Wa[k0 + 1];
            b.x = Wc[k0 * (2 * DIM) + ncol];
            b.y = Wc[(k0 + 1) * (2 * DIM) + ncol];
            acc = __builtin_amdgcn_wmma_f32_16x16x4_f32(
                false, a, false, b, (short)0, acc, false, false);
        }
        #pragma unroll
        for (int i = 0; i < 8; ++i)
            p[(i + 8 * half) * (2 * DIM) + ncol] = acc[i];
    }
}

// ---------------------------------------------------------------------------
// Kernel 2: v[b,e] = mm[b,e]*p[b,e] + p[b,512+e];  const[b] = sum_d u*bc + ba
// ---------------------------------------------------------------------------
__global__ __launch_bounds__(256) void prep_v(
    const float* __restrict__ mm, const float* __restrict__ p,
    const float* __restrict__ c1, const float* __restrict__ Wa,
    const float* __restrict__ bc, const float* __restrict__ ba,
    float* __restrict__ v, float* __restrict__ cconst)
{
    const int b = blockIdx.x;
    const int t = threadIdx.x;
    __shared__ float red[256];
    float part = 0.f;
    for (int d = t; d < DIM; d += 256) {
        const float u = c1[b * DIM + d] * Wa[d];
        part += u * bc[d];
        v[b * DIM + d] = mm[b * DIM + d] * p[b * 2 * DIM + d]
                       + p[b * 2 * DIM + DIM + d];
    }
    const float s = block_reduce_sum_256(part, red);
    if (t == 0) cconst[b] = s + ba[0];
}

// ---------------------------------------------------------------------------
// Kernel 3: logit[b,n] = sum_e v[b,e]*k[b,e,n] + const[b]  (streams all of k)
// 64 threads * float4 = 256 n per block, 16 blocks per batch. Coalesced b128.
// ---------------------------------------------------------------------------
__global__ __launch_bounds__(64) void logits_kernel(
    const float* __restrict__ k, const float* __restrict__ v,
    const float* __restrict__ cconst, float* __restrict__ logit)
{
    const int b = blockIdx.y;
    const int n = blockIdx.x * 256 + threadIdx.x * 4;
    const float* kb = k + (size_t)b * DIM * NN;
    const float* vb = v + b * DIM;
    float ax = 0.f, ay = 0.f, az = 0.f, aw = 0.f;
    #pragma unroll 4
    for (int e = 0; e < DIM; ++e) {
        const float  ve = vb[e];
        const float4 kv = *reinterpret_cast<const float4*>(kb + (size_t)e * NN + n);
        ax += ve * kv.x; ay += ve * kv.y; az += ve * kv.z; aw += ve * kv.w;
    }
    const float cc = cconst[b];
    float4 out; out.x = ax + cc; out.y = ay + cc; out.z = az + cc; out.w = aw + cc;
    *reinterpret_cast<float4*>(logit + b * NN + n) = out;
}

// ---------------------------------------------------------------------------
// Kernel 4: entmax bisection over N per row. alpha=1.5 => x=0.5*logit, power=2.
// One workgroup per row; row lives in LDS (16 KB << 320 KB/WGP).
// ---------------------------------------------------------------------------
__global__ __launch_bounds__(256) void entmax_kernel(
    const float* __restrict__ logit, float* __restrict__ wout)
{
    const int b = blockIdx.x;
    const int t = threadIdx.x;
    __shared__ float xs[NN];
    __shared__ float red[256];
    const float* lrow = logit + b * NN;
    for (int i = t; i < NN; i += 256) xs[i] = 0.5f * lrow[i];
    __syncthreads();

    float mloc = -3.402823466e38f;
    for (int i = t; i < NN; i += 256) mloc = fmaxf(mloc, xs[i]);
    red[t] = mloc;
    __syncthreads();
    for (int s = 128; s > 0; s >>= 1) {
        if (t < s) red[t] = fmaxf(red[t], red[t + s]);
        __syncthreads();
    }
    const float mx = red[0];
    __syncthreads();

    float tau_lo = mx - 1.0f;
    const float tau_hi = mx - 0.015625f;          // 4096^(1-1.5)
    float part = 0.f;
    for (int i = t; i < NN; i += 256) {
        const float z = fmaxf(xs[i] - tau_lo, 0.f);
        part += z * z;
    }
    const float f_lo = block_reduce_sum_256(part, red) - 1.0f;
    float dm = tau_hi - tau_lo;

    for (int it = 0; it < 50; ++it) {
        dm *= 0.5f;
        const float tau_m = tau_lo + dm;
        part = 0.f;
        for (int i = t; i < NN; i += 256) {
            const float z = fmaxf(xs[i] - tau_m, 0.f);
            part += z * z;
        }
        const float f_m = block_reduce_sum_256(part, red) - 1.0f;
        if (f_m * f_lo >= 0.f) tau_lo = tau_m;    // identical across threads
    }

    const float tau = tau_lo + dm * 0.5f;
    part = 0.f;
    for (int i = t; i < NN; i += 256) {
        const float z  = fmaxf(xs[i] - tau, 0.f);
        const float pv = z * z;
        xs[i] = pv;
        part += pv;
    }
    const float S   = block_reduce_sum_256(part, red);
    const float inv = 1.0f / S;
    for (int i = t; i < NN; i += 256) wout[b * NN + i] = xs[i] * inv;
}

// ---------------------------------------------------------------------------
// Kernel 5: r[b,d] = sum_n w[b,n]*k[b,d,n]. Second streaming pass over k
// (L2-resident candidate: 134 MB < 192 MB). Weights cached once in LDS and
// reused for 16 d-rows per block.
// ---------------------------------------------------------------------------
__global__ __launch_bounds__(256) void r_kernel(
    const float* __restrict__ k, const float* __restrict__ w,
    float* __restrict__ r)
{
    const int b  = blockIdx.y;
    const int dg = blockIdx.x;
    const int t  = threadIdx.x;
    __shared__ float wsh[NN];
    __shared__ float red[256];
    const float* wrow = w + b * NN;
    for (int i = t; i < NN; i += 256) wsh[i] = wrow[i];
    __syncthreads();
    const float* kb = k + (size_t)b * DIM * NN;
    for (int dd = 0; dd < 16; ++dd) {
        const int d = dg * 16 + dd;
        const float* krow = kb + (size_t)d * NN;
        float part = 0.f;
        #pragma unroll
        for (int i = 0; i < 4; ++i) {
            const int n = i * 1024 + t * 4;
            const float4 kv = *reinterpret_cast<const float4*>(krow + n);
            part += kv.x * wsh[n] + kv.y * wsh[n + 1]
                  + kv.z * wsh[n + 2] + kv.w * wsh[n + 3];
        }
        const float s = block_reduce_sum_256(part, red);
        if (t == 0) r[b * DIM + d] = s;
    }
}

// ---------------------------------------------------------------------------
extern "C" void kernel_launch(void* const* d_in, const int* in_sizes, int n_in,
                              void* d_out, int out_size, void* d_ws, size_t ws_size,
                              hipStream_t stream) {
    const float* m  = (const float*)d_in[0];   // [2,16,512]
    const float* k  = (const float*)d_in[1];   // [16,512,4096]
    const float* c  = (const float*)d_in[2];   // [2,16,512]
    const float* Wm = (const float*)d_in[3];   // [512,512]
    const float* bm = (const float*)d_in[4];   // [512]
    const float* Wc = (const float*)d_in[5];   // [512,1024]
    const float* bc = (const float*)d_in[6];   // [512]
    const float* Wa = (const float*)d_in[7];   // [1,512]
    const float* ba = (const float*)d_in[8];   // [1]
    const float* m1 = m + NB * DIM;            // m[-1]
    const float* c1 = c + NB * DIM;            // c[-1]

    float* ws     = (float*)d_ws;
    float* logit  = ws;                        // 16*4096
    float* mm     = logit + NB * NN;           // 16*512
    float* pbuf   = mm + NB * DIM;             // 16*1024
    float* vbuf   = pbuf + NB * 2 * DIM;       // 16*512
    float* cconst = vbuf + NB * DIM;           // 16   (~385 KB total)

    float* r_out = (float*)d_out;              // [16,512]
    float* w_out = (float*)d_out + NB * DIM;   // [16,1,4096]

    gemm_small_wmma<<<96, 32, 0, stream>>>(m1, Wm, bm, c1, Wa, Wc, mm, pbuf);
    prep_v<<<NB, 256, 0, stream>>>(mm, pbuf, c1, Wa, bc, ba, vbuf, cconst);
    logits_kernel<<<dim3(16, NB), 64, 0, stream>>>(k, vbuf, cconst, logit);
    entmax_kernel<<<NB, 256, 0, stream>>>(logit, w_out);
    r_kernel<<<dim3(32, NB), 256, 0, stream>>>(k, w_out, r_out);
}